// HeteGATMulti_38852274160087
// MI455X (gfx1250) — compile-verified
//
#include <hip/hip_runtime.h>
#include <cstdint>

#define P_   3
#define N_   4096
#define F_   1024
#define H_   8
#define D_   8
#define EMB_ 64
#define ATT_ 128
#define C_   16

typedef float v2f __attribute__((ext_vector_type(2)));
typedef float v8f __attribute__((ext_vector_type(8)));

// ---------------------------------------------------------------------------
// Generic fp32 WMMA GEMM: C = A[M,K] * B[K,N] (+ per-col bias, optional tanh).
// One wave per 16x(16*NB) tile, K multiple of 4, V_WMMA_F32_16X16X4_F32.
// A frag (16x4 f32): lanes 0-15 hold M=lane,K={k0,k0+1}; lanes 16-31 K={k0+2,+3}.
// Register-blocked over NB column tiles: A fragment loaded once per k-step.
// C/D frag: VGPR r -> rows r (lanes 0-15) and r+8 (lanes 16-31), col = lane%16.
// ---------------------------------------------------------------------------
template <int NB>
__global__ void wmma_gemm_f32(const float* __restrict__ A,
                              const float* __restrict__ B,
                              float* __restrict__ Cc,
                              int K, int lda, int ldb, int ldc,
                              long sA, long sB, long sC,
                              const float* __restrict__ bias, int act)
{
    const float* Ab = A + (long)blockIdx.z * sA;
    const float* Bb = B + (long)blockIdx.z * sB;
    float*       Cb = Cc + (long)blockIdx.z * sC;
    int m0 = blockIdx.x * 16, n0 = blockIdx.y * 16 * NB;
    int lane = threadIdx.x & 31;
    int r    = lane & 15;          // M row (A) / column (B,C)
    int g    = lane >> 4;          // lane-half selects K pair / row half
    int kk   = g * 2;
    v8f acc[NB] = {};
    for (int k0 = 0; k0 < K; k0 += 4) {
        v2f a;
        const float* Ap = Ab + (size_t)(m0 + r) * lda + (k0 + kk);
        a.x = Ap[0];
        a.y = Ap[1];
        const float* Bp = Bb + (size_t)(k0 + kk) * ldb + n0 + r;
        #pragma unroll
        for (int t = 0; t < NB; ++t) {
            v2f b;
            b.x = Bp[t * 16];
            b.y = Bp[t * 16 + ldb];
            acc[t] = __builtin_amdgcn_wmma_f32_16x16x4_f32(false, a, false, b,
                                                           (short)0, acc[t],
                                                           false, false);
        }
    }
    #pragma unroll
    for (int t = 0; t < NB; ++t) {
        int col = n0 + t * 16 + r;
        float bv = bias ? bias[col] : 0.0f;
        #pragma unroll
        for (int i = 0; i < 8; ++i) {
            int row = m0 + i + g * 8;
            float v = acc[t][i] + bv;
            if (act == 1) v = tanhf(v);
            Cb[(size_t)row * ldc + col] = v;
        }
    }
}

// ---------------------------------------------------------------------------
// Repack W [P,H,F,D] -> Wpacked [P,F,H*D] so the projection is a plain GEMM.
// ---------------------------------------------------------------------------
__global__ void repack_W(const float* __restrict__ W, float* __restrict__ Wp)
{
    int idx = blockIdx.x * blockDim.x + threadIdx.x;   // over P*F*EMB
    if (idx >= P_ * F_ * EMB_) return;
    int c = idx % EMB_;
    int f = (idx / EMB_) % F_;
    int p = idx / (EMB_ * F_);
    int h = c >> 3, d = c & 7;
    Wp[idx] = W[(((size_t)p * H_ + h) * F_ + f) * D_ + d];
}

// ---------------------------------------------------------------------------
// f1/f2 attention projections + head-major copy of fts for the LDS stage.
// ---------------------------------------------------------------------------
__global__ void f1f2_kernel(const float* __restrict__ fts,
                            const float* __restrict__ a1w, const float* __restrict__ a1b,
                            const float* __restrict__ a2w, const float* __restrict__ a2b,
                            float* __restrict__ f1, float* __restrict__ f2,
                            float* __restrict__ fts_h)
{
    int idx = blockIdx.x * blockDim.x + threadIdx.x;   // over P*H*N
    if (idx >= P_ * H_ * N_) return;
    int n = idx % N_;
    int h = (idx / N_) % H_;
    int p = idx / (N_ * H_);
    int ph = p * H_ + h;
    const float* fr = fts + ((size_t)p * N_ + n) * EMB_ + h * D_;
    float* fh = fts_h + ((size_t)ph * N_ + n) * D_;
    float s1 = a1b[ph], s2 = a2b[ph];
    #pragma unroll
    for (int d = 0; d < D_; ++d) {
        float v = fr[d];
        fh[d] = v;
        s1 += v * a1w[ph * D_ + d];
        s2 += v * a2w[ph * D_ + d];
    }
    f1[(size_t)ph * N_ + n] = s1;
    f2[(size_t)ph * N_ + n] = s2;
}

// ---------------------------------------------------------------------------
// gfx1250 async memory->LDS copy (ASYNCcnt-tracked, no VGPR round trip).
// dst_lds: wave-relative LDS byte address; src: 16B-aligned global address.
// ---------------------------------------------------------------------------
__device__ __forceinline__ void async_copy_b128(uint32_t dst_lds, const void* src)
{
    uint64_t gaddr = (uint64_t)(uintptr_t)src;
    asm volatile("global_load_async_to_lds_b128 %0, %1, off"
                 :: "v"(dst_lds), "v"(gaddr) : "memory");
}
__device__ __forceinline__ void async_wait0()
{
    asm volatile("s_wait_asynccnt 0x0" ::: "memory");
}

// ---------------------------------------------------------------------------
// Fused attention: streaming (online) softmax over m with fused coefs@fts.
// Block = 256 threads (8 waves); each wave owns one output row n.
// fts (head-major) + f2 tiles staged in LDS via global_load_async_to_lds_b128
// (stride-12 row pad keeps rows 16B-aligned and banks mostly conflict-free);
// bias_mat row streamed from L2 with global_prefetch ahead of the read.
// ---------------------------------------------------------------------------
#define CHUNK 1024
#define SSTR  12
__global__ void __launch_bounds__(256)
attn_kernel(const float* __restrict__ f1,
            const float* __restrict__ f2,
            const float* __restrict__ bias_mat,
            const float* __restrict__ fts_h,
            const float* __restrict__ h_bias,
            float* __restrict__ multi)
{
    __shared__ float s_fts[CHUNK * SSTR];   // 48 KB
    __shared__ float s_f2[CHUNK];           //  4 KB
    int p = blockIdx.z, h = blockIdx.y;
    int wave = threadIdx.x >> 5, lane = threadIdx.x & 31;
    int n  = blockIdx.x * 8 + wave;
    int ph = p * H_ + h;
    float f1n = f1[(size_t)ph * N_ + n];
    const float* brow = bias_mat + ((size_t)p * N_ + n) * N_;

    uint32_t lds_fts = (uint32_t)(uintptr_t)&s_fts[0];
    uint32_t lds_f2  = (uint32_t)(uintptr_t)&s_f2[0];

    float mx = -3.0e38f, ssum = 0.0f;
    float acc[8] = {0, 0, 0, 0, 0, 0, 0, 0};

    for (int c = 0; c < N_ / CHUNK; ++c) {
        __syncthreads();
        // ---- async stage: fts chunk (CHUNK rows x 8 floats, as 2 b128/row)
        const float* src = fts_h + ((size_t)ph * N_ + c * CHUNK) * D_;
        for (int i = threadIdx.x; i < CHUNK * 2; i += 256) {
            int row = i >> 1, part = i & 1;                 // 4-float half-rows
            async_copy_b128(lds_fts + (row * SSTR + part * 4) * 4,
                            src + row * D_ + part * 4);
        }
        // ---- async stage: f2 chunk (CHUNK floats as b128)
        const float* f2src = f2 + (size_t)ph * N_ + c * CHUNK;
        for (int i = threadIdx.x; i < CHUNK / 4; i += 256)
            async_copy_b128(lds_f2 + i * 16, f2src + i * 4);
        async_wait0();
        __syncthreads();

        const float* brow_c = brow + c * CHUNK;
        for (int it = 0; it < CHUNK; it += 32) {
            int m = it + lane;
            __builtin_prefetch(brow_c + m + 256, 0, 0);   // global_prefetch_b8
            float s = f1n + s_f2[m];
            s = (s > 0.0f) ? s : 0.01f * s;               // leaky_relu(0.01)
            s += brow_c[m];
            const float* fm = &s_fts[m * SSTR];
            if (s > mx) {                                  // rescale branch
                float sc = __expf(mx - s);
                ssum = ssum * sc + 1.0f;
                #pragma unroll
                for (int d = 0; d < 8; ++d) acc[d] = acc[d] * sc + fm[d];
                mx = s;
            } else {
                float pr = __expf(s - mx);
                ssum += pr;
                #pragma unroll
                for (int d = 0; d < 8; ++d) acc[d] += pr * fm[d];
            }
        }
    }

    // wave32 butterfly merge of softmax state (mx, ssum, acc[8])
    #pragma unroll
    for (int off = 16; off >= 1; off >>= 1) {
        float omx  = __shfl_xor(mx, off);
        float osum = __shfl_xor(ssum, off);
        float oacc[8];
        #pragma unroll
        for (int d = 0; d < 8; ++d) oacc[d] = __shfl_xor(acc[d], off);
        float nm = fmaxf(mx, omx);
        float sa = __expf(mx - nm), sb = __expf(omx - nm);
        ssum = ssum * sa + osum * sb;
        #pragma unroll
        for (int d = 0; d < 8; ++d) acc[d] = acc[d] * sa + oacc[d] * sb;
        mx = nm;
    }

    if (lane == 0) {
        float inv = 1.0f / ssum;
        #pragma unroll
        for (int d = 0; d < 8; ++d) {
            float v = acc[d] * inv + h_bias[ph * D_ + d];
            v = (v > 0.0f) ? v : (__expf(v) - 1.0f);       // elu
            multi[((size_t)n * P_ + p) * EMB_ + h * D_ + d] = v;
        }
    }
}

// ---------------------------------------------------------------------------
// Semantic (meta-path) attention: scores = v@u, softmax over P, combine.
// ---------------------------------------------------------------------------
__global__ void semantic_kernel(const float* __restrict__ vbuf,
                                const float* __restrict__ u,
                                const float* __restrict__ multi,
                                float* __restrict__ finalbuf,
                                float* __restrict__ out)
{
    int wave = threadIdx.x >> 5, lane = threadIdx.x & 31;
    int n = blockIdx.x * 8 + wave;
    const float* vr = vbuf + (size_t)n * P_ * ATT_;
    float s[P_];
    #pragma unroll
    for (int p = 0; p < P_; ++p) {
        float partial = 0.0f;
        for (int a = lane; a < ATT_; a += 32)
            partial += vr[p * ATT_ + a] * u[a];
        s[p] = partial;
    }
    #pragma unroll
    for (int off = 16; off >= 1; off >>= 1) {
        #pragma unroll
        for (int p = 0; p < P_; ++p) s[p] += __shfl_xor(s[p], off);
    }
    float mmax = fmaxf(s[0], fmaxf(s[1], s[2]));
    float e0 = __expf(s[0] - mmax), e1 = __expf(s[1] - mmax), e2 = __expf(s[2] - mmax);
    float inv = 1.0f / (e0 + e1 + e2);
    float a0 = e0 * inv, a1 = e1 * inv, a2 = e2 * inv;
    const float* mr = multi + (size_t)n * P_ * EMB_;
    for (int e = lane; e < EMB_; e += 32) {
        float fv = a0 * mr[e] + a1 * mr[EMB_ + e] + a2 * mr[2 * EMB_ + e];
        finalbuf[(size_t)n * EMB_ + e] = fv;
        out[(size_t)N_ * C_ + (size_t)n * EMB_ + e] = fv;     // 'final' output
    }
    if (lane == 0) {
        size_t base = (size_t)N_ * C_ + (size_t)N_ * EMB_ + (size_t)n * P_;
        out[base + 0] = a0;
        out[base + 1] = a1;
        out[base + 2] = a2;                                    // 'alphas' output
    }
}

// ---------------------------------------------------------------------------
extern "C" void kernel_launch(void* const* d_in, const int* in_sizes, int n_in,
                              void* d_out, int out_size, void* d_ws, size_t ws_size,
                              hipStream_t stream)
{
    const float* inputs   = (const float*)d_in[0];   // [P,N,F]
    const float* bias_mat = (const float*)d_in[1];   // [P,N,N]
    const float* W        = (const float*)d_in[2];   // [P,H,F,D]
    const float* a1w      = (const float*)d_in[3];
    const float* a1b      = (const float*)d_in[4];
    const float* a2w      = (const float*)d_in[5];
    const float* a2b      = (const float*)d_in[6];
    const float* hb       = (const float*)d_in[7];
    const float* Wsm      = (const float*)d_in[8];   // [64,128]
    const float* bs       = (const float*)d_in[9];
    const float* u        = (const float*)d_in[10];
    const float* Wc       = (const float*)d_in[11];  // [64,16]
    const float* bcv      = (const float*)d_in[12];
    float* out = (float*)d_out;
    float* ws  = (float*)d_ws;

    size_t o = 0;
    float* wpack    = ws + o; o += (size_t)P_ * F_ * EMB_;      // 196608
    float* fts      = ws + o; o += (size_t)P_ * N_ * EMB_;      // 786432
    float* fts_h    = ws + o; o += (size_t)P_ * H_ * N_ * D_;   // 786432
    float* f1       = ws + o; o += (size_t)P_ * H_ * N_;        // 98304
    float* f2      = ws + o; o += (size_t)P_ * H_ * N_;         // 98304
    float* multi    = ws + o; o += (size_t)N_ * P_ * EMB_;      // 786432
    float* vbuf     = ws + o; o += (size_t)N_ * P_ * ATT_;      // 1572864
    float* finalbuf = ws + o; o += (size_t)N_ * EMB_;           // 262144

    // 1) pack W -> [P,F,64]
    repack_W<<<dim3((P_ * F_ * EMB_ + 255) / 256), dim3(256), 0, stream>>>(W, wpack);

    // 2) projection GEMM, batched over P: fts[p] = inputs[p] @ wpack[p]
    wmma_gemm_f32<4><<<dim3(N_ / 16, EMB_ / 64, P_), dim3(32), 0, stream>>>(
        inputs, wpack, fts, F_, F_, EMB_, EMB_,
        (long)N_ * F_, (long)F_ * EMB_, (long)N_ * EMB_, nullptr, 0);

    // 3) f1/f2 + head-major fts
    f1f2_kernel<<<dim3((P_ * H_ * N_ + 255) / 256), dim3(256), 0, stream>>>(
        fts, a1w, a1b, a2w, a2b, f1, f2, fts_h);

    // 4) fused softmax-attention + aggregation (p slowest -> bias_mat L2 reuse)
    attn_kernel<<<dim3(N_ / 8, H_, P_), dim3(256), 0, stream>>>(
        f1, f2, bias_mat, fts_h, hb, multi);

    // 5) v = tanh(multi @ Ws + bs)   (M = N*P = 12288, K = 64, N = 128)
    wmma_gemm_f32<4><<<dim3((N_ * P_) / 16, ATT_ / 64, 1), dim3(32), 0, stream>>>(
        multi, Wsm, vbuf, EMB_, EMB_, ATT_, ATT_, 0, 0, 0, bs, 1);

    // 6) semantic softmax + combine -> final, alphas
    semantic_kernel<<<dim3(N_ / 8), dim3(256), 0, stream>>>(vbuf, u, multi, finalbuf, out);

    // 7) logits = final @ Wc + bc  -> d_out[0 : N*C]
    wmma_gemm_f32<1><<<dim3(N_ / 16, C_ / 16, 1), dim3(32), 0, stream>>>(
        finalbuf, Wc, out, EMB_, EMB_, C_, C_, 0, 0, 0, bcv, 0);
}